// Model_89704686944644
// MI455X (gfx1250) — compile-verified
//
#include <hip/hip_runtime.h>
#include <hip/hip_bf16.h>

typedef float v2f __attribute__((ext_vector_type(2)));
typedef float v8f __attribute__((ext_vector_type(8)));

#define ST_MAX_STEPS 64
#define ST_MAX_DIST  50.0f
#define ST_EPSILON   0.001f
#define ST_NSPH      128
#define ST_NTILES    8   // 128 spheres / 16 per WMMA column tile

// DPP16 control codes (stay within each 16-lane row; halves never mix)
#define DPP_XOR1  0xB1   // quad_perm [1,0,3,2]
#define DPP_XOR2  0x4E   // quad_perm [2,3,0,1]
#define DPP_MIR8  0x141  // row_half_mirror (lane ^ 7 inside each 8-group)
#define DPP_MIR16 0x140  // row_mirror      (lane ^ 15 inside each 16-row)

// min with DPP-permuted partner: VALU-only cross-lane reduction (no LDS pipe).
// CTRL must be a compile-time constant for the builtin.
template <int CTRL>
__device__ __forceinline__ float dpp_min(float x) {
    const int xi = __float_as_int(x);
    const int yi = __builtin_amdgcn_update_dpp(xi, xi, CTRL, 0xF, 0xF, true);
    return fminf(x, __int_as_float(yi));
}

// One wave32 traces 16 rays. A-matrix rows = rays (M), B columns = spheres (N).
// D = WMMA_F32_16x16x4( [px py pz pp], [-2cx -2cy -2cz 1] ) = |p|^2 - 2 p.c
// dist^2 = D + |c|^2 (per-lane const), dist = sqrt(dist^2) - radius (per-lane const).
__global__ __launch_bounds__(256) void sphere_trace_wmma_kernel(
    const float* __restrict__ ro, const float* __restrict__ rd,
    const float* __restrict__ sc, const float* __restrict__ sr,
    float* __restrict__ out, int R)
{
    const int lane = threadIdx.x & 31;
    const int m    = lane & 15;   // A row / B-D column owned by this lane
    const int h    = lane >> 4;   // wave half (selects K pairs 0,1 vs 2,3)
    const int wave = (int)((blockIdx.x * blockDim.x + threadIdx.x) >> 5);
    const int ray0 = wave * 16;

    int  ray   = ray0 + m;
    const bool owner = (h == 0) && (ray < R);
    if (ray >= R) ray = R - 1;     // clamp: keep all 32 lanes active (WMMA needs EXEC==~0)

    // ---- per-lane ray data (ray index m of this wave's tile) ----
    const float ox = ro[ray * 3 + 0], oy = ro[ray * 3 + 1], oz = ro[ray * 3 + 2];
    const float dx = rd[ray * 3 + 0], dy = rd[ray * 3 + 1], dz = rd[ray * 3 + 2];

    // ---- sphere tiles: B matrices + per-column constants (loop-invariant) ----
    v2f   Bt[ST_NTILES];
    float cct[ST_NTILES];
    float radt[ST_NTILES];
#pragma unroll
    for (int ti = 0; ti < ST_NTILES; ++ti) {
        const int s = ti * 16 + m;                 // sphere = column N of tile ti
        const float cx = sc[s * 3 + 0];
        const float cy = sc[s * 3 + 1];
        const float cz = sc[s * 3 + 2];
        cct[ti]  = fmaf(cx, cx, fmaf(cy, cy, cz * cz));   // |c|^2
        radt[ti] = sr[s];
        v2f b;                                    // B 4x16: h0 lanes K=0,1 ; h1 lanes K=2,3
        b.x = h ? (-2.0f * cz) : (-2.0f * cx);
        b.y = h ? 1.0f         : (-2.0f * cy);
        Bt[ti] = b;
    }

    float tc = 0.0f;                               // march parameter for ray m
    for (int step = 0; step < ST_MAX_STEPS; ++step) {
        const float px = fmaf(tc, dx, ox);
        const float py = fmaf(tc, dy, oy);
        const float pz = fmaf(tc, dz, oz);
        const float pp = fmaf(px, px, fmaf(py, py, pz * pz));

        v2f A;                                     // A 16x4: h0 lanes K=0,1 ; h1 lanes K=2,3
        A.x = h ? pz : px;
        A.y = h ? pp : py;

        float md[8];
#pragma unroll
        for (int v = 0; v < 8; ++v) md[v] = 1e30f;

#pragma unroll
        for (int ti = 0; ti < ST_NTILES; ++ti) {
            v8f c = {};                            // C = 0 (folds to inline constant)
            v8f d = __builtin_amdgcn_wmma_f32_16x16x4_f32(
                        false, A, false, Bt[ti], (short)0, c, false, false);
#pragma unroll
            for (int v = 0; v < 8; ++v) {
                float d2 = d[v] + cct[ti];         // |p-c|^2 (GEMM form)
                d2 = fmaxf(d2, 0.0f);              // rounding guard near surface
                const float dist = __builtin_amdgcn_sqrtf(d2) - radt[ti];
                md[v] = fminf(md[v], dist);        // min across tiles, elementwise rows
            }
        }

        // min across the 16 columns (spheres): 4-stage DPP16 butterfly, VALU-only.
        // Each stage's pairing stays inside the 16-lane row, so halves never mix.
#pragma unroll
        for (int v = 0; v < 8; ++v) {
            float x = md[v];
            x = dpp_min<DPP_XOR1>(x);   // pairs
            x = dpp_min<DPP_XOR2>(x);   // quads
            x = dpp_min<DPP_MIR8>(x);   // 8-groups
            x = dpp_min<DPP_MIR16>(x);  // full 16-row
            md[v] = x;
        }

        // deliver row (ray) m to this lane: binary select over md[k], k = m&7,
        // then one cross-half swap for rows owned by the other half.
        const int k  = m & 7;
        const float a0 = (k & 1) ? md[1] : md[0];
        const float a1 = (k & 1) ? md[3] : md[2];
        const float a2 = (k & 1) ? md[5] : md[4];
        const float a3 = (k & 1) ? md[7] : md[6];
        const float b0 = (k & 2) ? a1 : a0;
        const float b1 = (k & 2) ? a3 : a2;
        const float own = (k & 4) ? b1 : b0;       // sdf of ray (8h + k)
        const float oth = __shfl_xor(own, 16, 32); // sdf of ray (8(1-h) + k)
        const float sdf = ((m >> 3) == h) ? own : oth;

        const bool still = (sdf > ST_EPSILON) && (tc < ST_MAX_DIST);
        tc += still ? sdf : 0.0f;

        if (__ballot(still) == 0) break;           // wave-uniform early exit
    }

    if (owner) out[ray] = tc;
}

extern "C" void kernel_launch(void* const* d_in, const int* in_sizes, int n_in,
                              void* d_out, int out_size, void* d_ws, size_t ws_size,
                              hipStream_t stream) {
    const float* ro = (const float*)d_in[0];   // (R,3)
    const float* rd = (const float*)d_in[1];   // (R,3)
    const float* sc = (const float*)d_in[2];   // (128,3)
    const float* sr = (const float*)d_in[3];   // (128,)
    float* out = (float*)d_out;

    const int R = in_sizes[0] / 3;
    const int waves  = (R + 15) / 16;          // 16 rays per wave32
    const int blocks = (waves + 7) / 8;        // 8 waves (256 threads) per block

    sphere_trace_wmma_kernel<<<blocks, 256, 0, stream>>>(ro, rd, sc, sr, out, R);
}